// DeepseekV2MoEGate_472446403325
// MI455X (gfx1250) — compile-verified
//
#include <hip/hip_runtime.h>
#include <hip/hip_bf16.h>
#include <math.h>

// ---------------------------------------------------------------------------
// DeepSeek-V2 MoE gate for MI455X (gfx1250, wave32).
//   logits[T,160] = X[T,5120] @ W^T[5120,160]  via V_WMMA_F32_16X16X4_F32
//   then softmax + group-limited top-3-groups / top-6-experts per token.
//
// GEMM fragments are loaded DIRECTLY from global memory as b128 per lane,
// exploiting the fact that WMMA's K contraction is invariant under any
// permutation of K as long as A and B place the same physical k into the
// same (vgpr, lane-half) slot. Both X and W are K-contiguous row-major, so
// no transpose and no LDS staging is needed for the matrix inputs.
// ---------------------------------------------------------------------------

typedef __attribute__((ext_vector_type(2))) float v2f;
typedef __attribute__((ext_vector_type(4))) float v4f;
typedef __attribute__((ext_vector_type(8))) float v8f;

#define HDIM        5120
#define NEXPERT     160
#define NGROUP      8
#define GSIZE       20      // 160 / 8
#define TOPKG       3
#define TOPK        6
#define RSCALE      16.0f

#define M_PER_BLOCK 32      // tokens per workgroup
#define WAVES_M     2
#define WAVES_N     10      // 10 * 16 = 160 experts
#define NTHREADS    (WAVES_M * WAVES_N * 32)   // 640
#define LDS_STRIDE  161     // 160 + 1 pad: 8*161 % 64 != 0 -> no bank alias

__global__ __launch_bounds__(NTHREADS)
void moe_gate_kernel(const float* __restrict__ x,   // [T, 5120]
                     const float* __restrict__ w,   // [160, 5120]
                     float* __restrict__ out,       // [T*6 idx | T*6 weight]
                     int T)
{
    __shared__ float lds[M_PER_BLOCK][LDS_STRIDE];

    const int tid   = threadIdx.x;
    const int wave  = tid >> 5;
    const int lane  = tid & 31;
    const int mtile = wave / WAVES_N;        // 0..1
    const int ntile = wave % WAVES_N;        // 0..9
    const int row   = lane & 15;             // M index (A) / N index (B)
    const int half  = lane >> 4;             // 0: k+0..3, 1: k+4..7

    const long block_m = (long)blockIdx.x * M_PER_BLOCK;

    // clamp token row for safety (T is a multiple of 32 in practice)
    long arow = block_m + mtile * 16 + row;
    if (arow > (long)T - 1) arow = (long)T - 1;

    const float* aPtr = x + arow * (long)HDIM + half * 4;
    const float* bPtr = w + (long)(ntile * 16 + row) * (long)HDIM + half * 4;

    v8f acc = {};   // 16x16 f32 accumulator tile

    // K loop: 5120 / 8 = 640 steps, 2 x global_load_b128 + 2 x WMMA each.
    #pragma unroll 4
    for (int k = 0; k < HDIM; k += 8) {
        v4f av = *(const v4f*)(aPtr + k);
        v4f bv = *(const v4f*)(bPtr + k);
        v2f a0 = { av.x, av.y };
        v2f b0 = { bv.x, bv.y };
        v2f a1 = { av.z, av.w };
        v2f b1 = { bv.z, bv.w };
        // slot map (both A and B): half0 -> {k,k+1}/{k+2,k+3},
        //                          half1 -> {k+4,k+5}/{k+6,k+7}
        acc = __builtin_amdgcn_wmma_f32_16x16x4_f32(
                  false, a0, false, b0, (short)0, acc, false, false);
        acc = __builtin_amdgcn_wmma_f32_16x16x4_f32(
                  false, a1, false, b1, (short)0, acc, false, false);
    }

    // C/D layout: vgpr r, lanes 0-15 -> M = r, lanes 16-31 -> M = r + 8.
    const int t0 = mtile * 16 + 8 * half;
    const int ecol = ntile * 16 + row;
    #pragma unroll
    for (int r = 0; r < 8; ++r)
        lds[t0 + r][ecol] = acc[r];

    __syncthreads();

    // ------------------ per-token softmax + grouped top-k ------------------
    if (tid < M_PER_BLOCK) {
        const long tok = block_m + tid;
        if (tok < (long)T) {
            const float* L = lds[tid];

            // softmax denominator over all 160 experts
            float mx = -3.402823466e+38f;
            #pragma unroll 8
            for (int e = 0; e < NEXPERT; ++e) mx = fmaxf(mx, L[e]);
            float sum = 0.0f;
            #pragma unroll 8
            for (int e = 0; e < NEXPERT; ++e) sum += expf(L[e] - mx);

            // group maxima on raw logits (softmax is monotone)
            float gm[NGROUP];
            #pragma unroll
            for (int g = 0; g < NGROUP; ++g) {
                float m = -3.402823466e+38f;
                #pragma unroll
                for (int j = 0; j < GSIZE; ++j)
                    m = fmaxf(m, L[g * GSIZE + j]);
                gm[g] = m;
            }

            // top-3 groups (ties -> lowest group index, like lax.top_k)
            bool gsel[NGROUP];
            #pragma unroll
            for (int g = 0; g < NGROUP; ++g) gsel[g] = false;
            #pragma unroll
            for (int it = 0; it < TOPKG; ++it) {
                int   best = 0;
                float bv   = -3.402823466e+38f;
                #pragma unroll
                for (int g = 0; g < NGROUP; ++g)
                    if (!gsel[g] && gm[g] > bv) { bv = gm[g]; best = g; }
                gsel[best] = true;
            }

            // top-6 experts among selected groups (60 candidates, all > 0
            // after softmax so masked-out zeros can never win)
            float bestv[TOPK];
            int   besti[TOPK];
            #pragma unroll
            for (int j = 0; j < TOPK; ++j) { bestv[j] = -3.402823466e+38f; besti[j] = 0; }
            for (int g = 0; g < NGROUP; ++g) {
                if (!gsel[g]) continue;
                #pragma unroll
                for (int j = 0; j < GSIZE; ++j) {
                    const int   idx = g * GSIZE + j;
                    const float v   = L[idx];
                    if (v > bestv[TOPK - 1]) {      // strict > keeps lowest idx on ties
                        int p = TOPK - 1;
                        while (p > 0 && v > bestv[p - 1]) {
                            bestv[p] = bestv[p - 1];
                            besti[p] = besti[p - 1];
                            --p;
                        }
                        bestv[p] = v;
                        besti[p] = idx;
                    }
                }
            }

            const float inv = 1.0f / sum;
            float* out_idx = out;
            float* out_wgt = out + (long)T * TOPK;
            #pragma unroll
            for (int j = 0; j < TOPK; ++j) {
                out_idx[tok * TOPK + j] = (float)besti[j];
                out_wgt[tok * TOPK + j] = expf(bestv[j] - mx) * inv * RSCALE;
            }
        }
    }
}

extern "C" void kernel_launch(void* const* d_in, const int* in_sizes, int n_in,
                              void* d_out, int out_size, void* d_ws, size_t ws_size,
                              hipStream_t stream)
{
    const float* x   = (const float*)d_in[0];   // [4,4096,5120] f32
    const float* wgt = (const float*)d_in[1];   // [160,5120]    f32
    float*       out = (float*)d_out;           // [T*6 idx | T*6 weight] f32

    const int T = in_sizes[0] / HDIM;           // 16384
    const int blocks = (T + M_PER_BLOCK - 1) / M_PER_BLOCK;  // 512

    moe_gate_kernel<<<blocks, NTHREADS, 0, stream>>>(x, wgt, out, T);
}